// GraphPolygonEncoder_9809705304971
// MI455X (gfx1250) — compile-verified
//
#include <hip/hip_runtime.h>
#include <math.h>

#define NNODE 50000
#define NEDGE 400000
#define NGRAPH 512

typedef __attribute__((ext_vector_type(2))) float v2f;
typedef __attribute__((ext_vector_type(8))) float v8f;

// ---------------------------------------------------------------------------
// Generic fp32 WMMA GEMM: C[M,Ncols] = A[M,K] @ B[K,Ncols] (+bias)(+relu)
// One wave (32 threads) computes a 16 x (16*NT) tile using V_WMMA_F32_16X16X4_F32.
// M must be a multiple of 16, K a multiple of 4, Ncols == 16*NT.
// mode: 0 = none, 1 = +bias, 2 = +bias then relu
// ---------------------------------------------------------------------------
template <int NT>
__global__ __launch_bounds__(32) void gemm_wmma_f32(
    const float* __restrict__ A, const float* __restrict__ B,
    const float* __restrict__ bias, float* __restrict__ C,
    int M, int K, int Ncols, int mode) {
  const int tile = blockIdx.x;        // 16-row tile index
  const int lane = threadIdx.x;       // 0..31
  const int half = lane >> 4;         // 0: lanes 0-15, 1: lanes 16-31
  const int r = lane & 15;

  v8f acc[NT];
#pragma unroll
  for (int t = 0; t < NT; ++t)
#pragma unroll
    for (int j = 0; j < 8; ++j) acc[t][j] = 0.0f;

  const int row = tile * 16 + r;
  const float* __restrict__ arow = A + (size_t)row * K;

  for (int k0 = 0; k0 < K; k0 += 4) {
    // A 16x4 tile layout: lanes 0-15 hold K=k0..k0+1, lanes 16-31 K=k0+2..k0+3
    v2f a;
    a.x = arow[k0 + 2 * half];
    a.y = arow[k0 + 2 * half + 1];
#pragma unroll
    for (int t = 0; t < NT; ++t) {
      const int col = t * 16 + r;
      // B 4x16 tile: VGPR0 lanes0-15 K=k0, lanes16-31 K=k0+2; VGPR1: +1
      v2f b;
      b.x = B[(size_t)(k0 + 2 * half) * Ncols + col];
      b.y = B[(size_t)(k0 + 2 * half + 1) * Ncols + col];
      acc[t] = __builtin_amdgcn_wmma_f32_16x16x4_f32(
          false, a, false, b, (short)0, acc[t], false, false);
    }
  }

#pragma unroll
  for (int t = 0; t < NT; ++t) {
    const int col = t * 16 + r;
    const float bv = (mode > 0) ? bias[col] : 0.0f;
#pragma unroll
    for (int j = 0; j < 8; ++j) {
      const int orow = tile * 16 + j + 8 * half;  // C VGPR j: M=j (+8 upper half)
      float v = acc[t][j] + bv;
      if (mode == 2) v = fmaxf(v, 0.0f);
      C[(size_t)orow * Ncols + col] = v;
    }
  }
}

// ---------------------------------------------------------------------------
// Fill a u32 buffer with a constant (used to init max/sum/acc buffers)
// ---------------------------------------------------------------------------
__global__ void fill_u32(unsigned* __restrict__ p, unsigned val, long n) {
  long i = (long)blockIdx.x * blockDim.x + threadIdx.x;
  if (i < n) p[i] = val;
}

// ---------------------------------------------------------------------------
// Edge pass 1: per (edge, head) logit + ordered-uint atomic max per (dst, head)
// ee = ea @ We recomputed inline (K=4, We is L2/cache resident)
// ---------------------------------------------------------------------------
__global__ __launch_bounds__(256) void edge_logit_kernel(
    const float* __restrict__ xl, const float* __restrict__ xr,
    const float* __restrict__ ea, const int* __restrict__ src,
    const int* __restrict__ dst, const float* __restrict__ We,
    const float* __restrict__ att, float* __restrict__ logit,
    unsigned* __restrict__ nmax, int H) {
  const long t = (long)blockIdx.x * blockDim.x + threadIdx.x;
  const long total = (long)NEDGE * H;
  if (t >= total) return;
  const int e = (int)(t / H);
  const int h = (int)(t % H);
  const int HC = H * 32;
  const int s = src[e];
  const int d = dst[e];
  const float* __restrict__ xls = xl + (size_t)s * HC + h * 32;
  const float* __restrict__ xrd = xr + (size_t)d * HC + h * 32;
  const float e0 = ea[e * 4 + 0], e1 = ea[e * 4 + 1];
  const float e2 = ea[e * 4 + 2], e3 = ea[e * 4 + 3];
  float lg = 0.0f;
#pragma unroll 8
  for (int c = 0; c < 32; ++c) {
    const int col = h * 32 + c;
    float m = xls[c] + xrd[c] +
              e0 * We[col] + e1 * We[HC + col] +
              e2 * We[2 * HC + col] + e3 * We[3 * HC + col];
    m = (m > 0.0f) ? m : 0.2f * m;  // leaky_relu(0.2)
    lg += m * att[col];
  }
  logit[t] = lg;
  // monotone float -> uint encoding for atomic max
  const int li = __float_as_int(lg);
  const unsigned enc = (li < 0) ? ~(unsigned)li : ((unsigned)li | 0x80000000u);
  atomicMax(&nmax[(size_t)d * H + h], enc);
}

// ---------------------------------------------------------------------------
// Edge pass 2: p = exp(logit - max[dst]); psum[dst,h] += p;
// acc[dst,h,:] += p * xl[src,h,:]   (normalization deferred to node kernel)
// ---------------------------------------------------------------------------
__global__ __launch_bounds__(256) void edge_accum_kernel(
    const float* __restrict__ xl, const int* __restrict__ src,
    const int* __restrict__ dst, const float* __restrict__ logit,
    const unsigned* __restrict__ nmax, float* __restrict__ psum,
    float* __restrict__ acc, int H) {
  const long t = (long)blockIdx.x * blockDim.x + threadIdx.x;
  const long total = (long)NEDGE * H;
  if (t >= total) return;
  const int e = (int)(t / H);
  const int h = (int)(t % H);
  const int HC = H * 32;
  const int s = src[e];
  const int d = dst[e];
  const unsigned enc = nmax[(size_t)d * H + h];
  const float mx = (enc & 0x80000000u) ? __int_as_float((int)(enc & 0x7fffffffu))
                                       : __int_as_float((int)~enc);
  const float p = __expf(logit[t] - mx);
  atomicAdd(&psum[(size_t)d * H + h], p);
  const float* __restrict__ xls = xl + (size_t)s * HC + h * 32;
  float* __restrict__ ac = acc + (size_t)d * HC + h * 32;
#pragma unroll 8
  for (int c = 0; c < 32; ++c) atomicAdd(&ac[c], p * xls[c]);
}

// ---------------------------------------------------------------------------
// Node finalize: one wave32 per node.
// out = gelu(layernorm(acc/psum + bias + res))
// OT = H*32 (concat layers) or 32 (last layer, H=1 mean == identity)
// ---------------------------------------------------------------------------
__global__ __launch_bounds__(256) void node_finalize_kernel(
    const float* __restrict__ acc, const float* __restrict__ psum,
    const float* __restrict__ res, const float* __restrict__ bias,
    const float* __restrict__ ln_g, const float* __restrict__ ln_b,
    float* __restrict__ out, int H, int OT) {
  const long wave = ((long)blockIdx.x * blockDim.x + threadIdx.x) >> 5;
  const int lane = threadIdx.x & 31;
  if (wave >= NNODE) return;
  const int PC = OT >> 5;  // channels per lane (4 or 1)
  float v[4];
  float s = 0.0f;
  for (int i = 0; i < PC; ++i) {
    const int c = i * 32 + lane;
    const float ps = psum[wave * H + (c >> 5)];
    const float a = acc[wave * OT + c];
    float o = (ps > 0.0f) ? (a / ps) : 0.0f;  // nodes w/o in-edges -> 0
    o += bias[c] + res[wave * OT + c];
    v[i] = o;
    s += o;
  }
#pragma unroll
  for (int off = 16; off; off >>= 1) s += __shfl_xor(s, off);
  const float mu = s / (float)OT;
  float var = 0.0f;
  for (int i = 0; i < PC; ++i) {
    const float dlt = v[i] - mu;
    var += dlt * dlt;
  }
#pragma unroll
  for (int off = 16; off; off >>= 1) var += __shfl_xor(var, off);
  const float inv = rsqrtf(var / (float)OT + 1e-5f);
  for (int i = 0; i < PC; ++i) {
    const int c = i * 32 + lane;
    float y = (v[i] - mu) * inv * ln_g[c] + ln_b[c];
    y = 0.5f * y * (1.0f + erff(y * 0.70710678118654752f));  // exact gelu
    out[wave * OT + c] = y;
  }
}

// ---------------------------------------------------------------------------
// Attention pooling: one wave32 per node; lane == channel (HID=32)
// pooled[batch[n],c] += h[n,c] * sigmoid(h[n,:]·attW + attb)
// ---------------------------------------------------------------------------
__global__ __launch_bounds__(256) void pool_kernel(
    const float* __restrict__ h, const int* __restrict__ batch,
    const float* __restrict__ attW, const float* __restrict__ attb,
    float* __restrict__ pooled) {
  const long wave = ((long)blockIdx.x * blockDim.x + threadIdx.x) >> 5;
  const int lane = threadIdx.x & 31;
  if (wave >= NNODE) return;
  const float v = h[wave * 32 + lane];
  float d = v * attW[lane];
#pragma unroll
  for (int off = 16; off; off >>= 1) d += __shfl_xor(d, off);
  const float sgm = 1.0f / (1.0f + __expf(-(d + attb[0])));
  const int g = batch[wave];
  atomicAdd(&pooled[(size_t)g * 32 + lane], v * sgm);
}

// ---------------------------------------------------------------------------
// Row L2-normalize: one wave32 per graph row of 128
// ---------------------------------------------------------------------------
__global__ __launch_bounds__(256) void l2norm_kernel(
    const float* __restrict__ emb, float* __restrict__ out) {
  const long wave = ((long)blockIdx.x * blockDim.x + threadIdx.x) >> 5;
  const int lane = threadIdx.x & 31;
  if (wave >= NGRAPH) return;
  float v[4];
  float ss = 0.0f;
#pragma unroll
  for (int i = 0; i < 4; ++i) {
    v[i] = emb[wave * 128 + i * 32 + lane];
    ss += v[i] * v[i];
  }
#pragma unroll
  for (int off = 16; off; off >>= 1) ss += __shfl_xor(ss, off);
  const float inv = 1.0f / fmaxf(sqrtf(ss), 1e-12f);
#pragma unroll
  for (int i = 0; i < 4; ++i) out[wave * 128 + i * 32 + lane] = v[i] * inv;
}

// ---------------------------------------------------------------------------
// Host orchestration
// ---------------------------------------------------------------------------
static inline int cdiv(long a, long b) { return (int)((a + b - 1) / b); }

extern "C" void kernel_launch(void* const* d_in, const int* in_sizes, int n_in,
                              void* d_out, int out_size, void* d_ws, size_t ws_size,
                              hipStream_t stream) {
  // Pytree-flattened inputs (alphabetical dict order, None leaves dropped):
  // 0:batch 1:edge_attr 2:edge_index 3:W1 4:W2 5:attW 6:attb 7:b1 8:b2 9:inW 10:inb
  // then layers[0..3]: We,Wl,Wr,att,bias,ln_b,ln_g[,resW,resb]; last: x
  const int*   batch = (const int*)d_in[0];
  const float* eattr = (const float*)d_in[1];
  const int*   ei    = (const int*)d_in[2];       // [2,E]: src row then dst row
  const float* W1    = (const float*)d_in[3];
  const float* W2    = (const float*)d_in[4];
  const float* attW  = (const float*)d_in[5];
  const float* attb  = (const float*)d_in[6];
  const float* b1    = (const float*)d_in[7];
  const float* b2    = (const float*)d_in[8];
  const float* inW   = (const float*)d_in[9];
  const float* inb   = (const float*)d_in[10];

  struct Layer { const float *We, *Wl, *Wr, *att, *bias, *ln_b, *ln_g, *resW, *resb; int H; };
  Layer L[4];
  const int   heads[4]  = {4, 4, 4, 1};
  const bool  hasres[4] = {true, false, false, true};
  int idx = 11;
  for (int l = 0; l < 4; ++l) {
    L[l].We   = (const float*)d_in[idx++];
    L[l].Wl   = (const float*)d_in[idx++];
    L[l].Wr   = (const float*)d_in[idx++];
    L[l].att  = (const float*)d_in[idx++];
    L[l].bias = (const float*)d_in[idx++];
    L[l].ln_b = (const float*)d_in[idx++];
    L[l].ln_g = (const float*)d_in[idx++];
    if (hasres[l]) {
      L[l].resW = (const float*)d_in[idx++];
      L[l].resb = (const float*)d_in[idx++];
    } else {
      L[l].resW = nullptr; L[l].resb = nullptr;
    }
    L[l].H = heads[l];
  }
  const float* x = (const float*)d_in[idx];  // [N,8]

  // ---- workspace layout (fp32 unless noted) ----
  float* ws = (float*)d_ws;
  const size_t NB = (size_t)NNODE * 128;
  float*    bufA   = ws;                 // h (ping)
  float*    bufB   = ws + NB;            // xl
  float*    bufC   = ws + 2 * NB;        // xr
  float*    bufD   = ws + 3 * NB;        // acc (softmax numerator)
  float*    bufE   = ws + 4 * NB;        // residual projection
  float*    bufF   = ws + 5 * NB;        // h (pong)
  float*    logit  = ws + 6 * NB;                         // [E, H<=4]
  unsigned* nmax   = (unsigned*)(logit + (size_t)NEDGE * 4);  // [N, H]
  float*    psum   = (float*)(nmax + (size_t)NNODE * 4);      // [N, H]
  float*    pooled = psum + (size_t)NNODE * 4;                // [G, 32]
  float*    zbuf   = pooled + (size_t)NGRAPH * 32;            // [G, 128]
  float*    ebuf   = zbuf + (size_t)NGRAPH * 128;             // [G, 128]

  const int* src = ei;
  const int* dst = ei + NEDGE;
  const int NTILES = NNODE / 16;  // 3125

  // ---- input projection: h = x @ inW + inb  (K=8 -> 2 WMMA steps) ----
  gemm_wmma_f32<2><<<NTILES, 32, 0, stream>>>(x, inW, inb, bufA, NNODE, 8, 32, 1);

  float* h = bufA;
  int in_ch = 32;
  for (int l = 0; l < 4; ++l) {
    const int H = L[l].H;
    const int HC = H * 32;            // == out_total for every layer here
    const long totEH = (long)NEDGE * H;

    // node transforms
    if (HC == 128) {
      gemm_wmma_f32<8><<<NTILES, 32, 0, stream>>>(h, L[l].Wl, nullptr, bufB, NNODE, in_ch, HC, 0);
      gemm_wmma_f32<8><<<NTILES, 32, 0, stream>>>(h, L[l].Wr, nullptr, bufC, NNODE, in_ch, HC, 0);
    } else {
      gemm_wmma_f32<2><<<NTILES, 32, 0, stream>>>(h, L[l].Wl, nullptr, bufB, NNODE, in_ch, HC, 0);
      gemm_wmma_f32<2><<<NTILES, 32, 0, stream>>>(h, L[l].Wr, nullptr, bufC, NNODE, in_ch, HC, 0);
    }

    // residual path
    const float* resp = h;
    if (L[l].resW) {
      if (HC == 128)
        gemm_wmma_f32<8><<<NTILES, 32, 0, stream>>>(h, L[l].resW, L[l].resb, bufE, NNODE, in_ch, HC, 1);
      else
        gemm_wmma_f32<2><<<NTILES, 32, 0, stream>>>(h, L[l].resW, L[l].resb, bufE, NNODE, in_ch, HC, 1);
      resp = bufE;
    }

    // init per-node max (= encoded -inf), sum, accumulator
    fill_u32<<<cdiv((long)NNODE * H, 256), 256, 0, stream>>>(nmax, 0x007fffffu, (long)NNODE * H);
    fill_u32<<<cdiv((long)NNODE * H, 256), 256, 0, stream>>>((unsigned*)psum, 0u, (long)NNODE * H);
    fill_u32<<<cdiv((long)NNODE * HC, 256), 256, 0, stream>>>((unsigned*)bufD, 0u, (long)NNODE * HC);

    // edge passes (segment softmax + weighted aggregation)
    edge_logit_kernel<<<cdiv(totEH, 256), 256, 0, stream>>>(
        bufB, bufC, eattr, src, dst, L[l].We, L[l].att, logit, nmax, H);
    edge_accum_kernel<<<cdiv(totEH, 256), 256, 0, stream>>>(
        bufB, src, dst, logit, nmax, psum, bufD, H);

    // finalize: normalize + bias + residual + layernorm + gelu
    float* hn = (h == bufA) ? bufF : bufA;
    node_finalize_kernel<<<cdiv((long)NNODE * 32, 256), 256, 0, stream>>>(
        bufD, psum, resp, L[l].bias, L[l].ln_g, L[l].ln_b, hn, H, HC);
    h = hn;
    in_ch = HC;
  }

  // ---- attention pooling over graphs ----
  fill_u32<<<cdiv((long)NGRAPH * 32, 256), 256, 0, stream>>>((unsigned*)pooled, 0u, (long)NGRAPH * 32);
  pool_kernel<<<cdiv((long)NNODE * 32, 256), 256, 0, stream>>>(h, batch, attW, attb, pooled);

  // ---- final MLP + L2 normalize ----
  gemm_wmma_f32<8><<<NGRAPH / 16, 32, 0, stream>>>(pooled, W1, b1, zbuf, NGRAPH, 32, 128, 2);
  gemm_wmma_f32<8><<<NGRAPH / 16, 32, 0, stream>>>(zbuf, W2, b2, ebuf, NGRAPH, 128, 128, 1);
  l2norm_kernel<<<cdiv((long)NGRAPH * 32, 256), 256, 0, stream>>>(ebuf, (float*)d_out);
}